// Loss_31207232373378
// MI455X (gfx1250) — compile-verified
//
#include <hip/hip_runtime.h>
#include <math.h>

// ---------------------------------------------------------------------------
// Bone-direction loss for MI455X (gfx1250, wave32).
// loss = 0.1 * ( mean|p*m - t*m| + 0.1 * mean((dir(p*m)-dir(t*m))*m)^2 )
// Bandwidth-bound: 157 MB read @ 23.3 TB/s => ~6.7us floor. Single streaming
// pass, LDS-staged bone gather, WMMA (f32 16x16x4, ones-A => exact column
// sums) for the cross-wave reduction.
// ---------------------------------------------------------------------------

typedef __attribute__((ext_vector_type(2))) float v2f;
typedef __attribute__((ext_vector_type(8))) float v8f;

#define TINY_F32 1.17549435e-38f   // jnp.finfo(float32).tiny == FLT_MIN
#define THREADS 256
#define ROW 150
#define NBONES 50
#define ROWS_PER_BLOCK 16
#define NROWS (128 * 1024)
#define NBLOCKS (NROWS / ROWS_PER_BLOCK)      // 8192
#define NTOT 19660800.0f                      // 128*1024*150

__device__ __forceinline__ float wave_reduce_add(float v) {
#pragma unroll
    for (int off = 16; off > 0; off >>= 1)
        v += __shfl_xor(v, off, 32);
    return v;
}

// part[0..7]  = per-wave L1 partials
// part[8..15] = per-wave squared-term partials
// Uses V_WMMA_F32_16X16X4_F32 with A == ones so D[m,n] = sum_k B[k,n] exactly.
// Columns 0,1 carry the 8 L1 partials (4 per column); columns 2,3 carry the
// 8 Sq partials. Executed by every wave (same data) so EXEC is all-ones.
__device__ __forceinline__ void wmma_finish(const float* part,
                                            float& totL1, float& totSq) {
    const int lane = threadIdx.x & 31;
    const int col  = lane & 15;
    const int half = lane >> 4;

    v2f a; a[0] = 1.0f; a[1] = 1.0f;     // A = ones(16x4)
    v2f b; b[0] = 0.0f; b[1] = 0.0f;
    if (col < 4) {
        // Each column n owns 4 B slots: {lane n, lane n+16} x {vgpr0, vgpr1}.
        // Exact k-slot mapping is irrelevant for a pure column sum.
        const int base = col * 4 + half * 2;
        b[0] = part[base];
        b[1] = part[base + 1];
    }
    v8f c = {};
    // (neg_a, A, neg_b, B, c_mod, C, reuse_a, reuse_b)
    c = __builtin_amdgcn_wmma_f32_16x16x4_f32(false, a, false, b,
                                              (short)0, c, false, false);
    // Row 0 of D lives in c[0], lanes 0..15 (rows replicated since A==ones).
    const float c0 = c[0];
    totL1 = __shfl(c0, 0, 32) + __shfl(c0, 1, 32);
    totSq = __shfl(c0, 2, 32) + __shfl(c0, 3, 32);
}

__global__ __launch_bounds__(THREADS)
void bone_loss_partial(const float* __restrict__ preds,
                       const float* __restrict__ targets,
                       float* __restrict__ partials) {
    __shared__ __align__(16) float Psh[ROWS_PER_BLOCK * ROW];  // masked preds
    __shared__ __align__(16) float Tsh[ROWS_PER_BLOCK * ROW];  // targets (t*m == t)
    __shared__ float part[16];

    const int tid = threadIdx.x;
    const size_t base = (size_t)blockIdx.x * (ROWS_PER_BLOCK * ROW);
    const float4* __restrict__ p4 = reinterpret_cast<const float4*>(preds + base);
    const float4* __restrict__ t4 = reinterpret_cast<const float4*>(targets + base);
    float4* Ps4 = reinterpret_cast<float4*>(Psh);
    float4* Ts4 = reinterpret_cast<float4*>(Tsh);

    // --- Streaming load (b128), mask application, L1 accumulation ---
    float accL1 = 0.0f;
    const int NV = ROWS_PER_BLOCK * ROW / 4;   // 600 float4 per input
    for (int i = tid; i < NV; i += THREADS) {
        const float4 tv = t4[i];
        const float4 pv = p4[i];
        float4 pm;
        pm.x = (tv.x != 0.0f) ? pv.x : 0.0f;
        pm.y = (tv.y != 0.0f) ? pv.y : 0.0f;
        pm.z = (tv.z != 0.0f) ? pv.z : 0.0f;
        pm.w = (tv.w != 0.0f) ? pv.w : 0.0f;
        accL1 += fabsf(pm.x - tv.x) + fabsf(pm.y - tv.y)
               + fabsf(pm.z - tv.z) + fabsf(pm.w - tv.w);
        Ps4[i] = pm;
        Ts4[i] = tv;
    }
    __syncthreads();

    // --- Bone direction phase: 16 rows x 50 bones out of LDS ---
    float accSq = 0.0f;
    for (int tsk = tid; tsk < ROWS_PER_BLOCK * NBONES; tsk += THREADS) {
        const int r    = tsk / NBONES;
        const int bone = tsk - r * NBONES;
        const int ro = r * ROW;
        const int ja = ro + bone * 3;                       // joint i
        const int jb = ro + ((bone + 1) % NBONES) * 3;      // joint (i+1)%50

        const float pdx = Psh[ja + 0] - Psh[jb + 0];
        const float pdy = Psh[ja + 1] - Psh[jb + 1];
        const float pdz = Psh[ja + 2] - Psh[jb + 2];
        const float tdx = Tsh[ja + 0] - Tsh[jb + 0];
        const float tdy = Tsh[ja + 1] - Tsh[jb + 1];
        const float tdz = Tsh[ja + 2] - Tsh[jb + 2];

        const float pinv = 1.0f / (sqrtf(pdx*pdx + pdy*pdy + pdz*pdz) + TINY_F32);
        const float tinv = 1.0f / (sqrtf(tdx*tdx + tdy*tdy + tdz*tdz) + TINY_F32);

        // mask[:, :, :150] element (bone i, axis c) == mask of joint i axis c
        const float m0 = (Tsh[ja + 0] != 0.0f) ? 1.0f : 0.0f;
        const float m1 = (Tsh[ja + 1] != 0.0f) ? 1.0f : 0.0f;
        const float m2 = (Tsh[ja + 2] != 0.0f) ? 1.0f : 0.0f;

        const float d0 = (pdx * pinv - tdx * tinv) * m0;
        const float d1 = (pdy * pinv - tdy * tinv) * m1;
        const float d2 = (pdz * pinv - tdz * tinv) * m2;
        accSq += d0*d0 + d1*d1 + d2*d2;
    }

    // --- Block reduction: shfl within wave, WMMA across the 8 waves ---
    accL1 = wave_reduce_add(accL1);
    accSq = wave_reduce_add(accSq);
    const int lane = tid & 31, w = tid >> 5;
    if (lane == 0) { part[w] = accL1; part[8 + w] = accSq; }
    __syncthreads();

    float totL1, totSq;
    wmma_finish(part, totL1, totSq);
    if (tid == 0) {
        partials[2 * blockIdx.x]     = totL1;
        partials[2 * blockIdx.x + 1] = totSq;
    }
}

__global__ __launch_bounds__(THREADS)
void bone_loss_final(const float* __restrict__ partials,
                     float* __restrict__ out) {
    __shared__ float part[16];
    const int tid = threadIdx.x;

    float accL1 = 0.0f, accSq = 0.0f;
    for (int i = tid; i < NBLOCKS; i += THREADS) {   // 32 iters/thread, fixed order
        accL1 += partials[2 * i];
        accSq += partials[2 * i + 1];
    }
    accL1 = wave_reduce_add(accL1);
    accSq = wave_reduce_add(accSq);
    const int lane = tid & 31, w = tid >> 5;
    if (lane == 0) { part[w] = accL1; part[8 + w] = accSq; }
    __syncthreads();

    float totL1, totSq;
    wmma_finish(part, totL1, totSq);
    if (tid == 0) {
        const float loss = totL1 / NTOT + 0.1f * (totSq / NTOT);
        out[0] = 0.1f * loss;
    }
}

extern "C" void kernel_launch(void* const* d_in, const int* in_sizes, int n_in,
                              void* d_out, int out_size, void* d_ws, size_t ws_size,
                              hipStream_t stream) {
    (void)in_sizes; (void)n_in; (void)out_size; (void)ws_size;
    const float* preds   = (const float*)d_in[0];
    const float* targets = (const float*)d_in[1];
    float* partials = (float*)d_ws;   // needs 8192*2*4 = 64 KB

    bone_loss_partial<<<NBLOCKS, THREADS, 0, stream>>>(preds, targets, partials);
    bone_loss_final<<<1, THREADS, 0, stream>>>(partials, (float*)d_out);
}